// PairwisePredictionHead_51814485459081
// MI455X (gfx1250) — compile-verified
//
#include <hip/hip_runtime.h>
#include <hip/hip_bf16.h>
#include <math.h>

// Problem constants: B=1, L=512, H=1024, C=2
#define LSEQ 512
#define HDIM 1024
#define HC   256   // h-chunk for pairwise stage
#define LDSP (HC + 4) // padded LDS row stride (floats): 260 % 64 == 4 -> conflict-free

typedef __attribute__((ext_vector_type(2))) float v2f;
typedef __attribute__((ext_vector_type(8))) float v8f;

__device__ __forceinline__ float fast_tanh(float x) {
#if __has_builtin(__builtin_amdgcn_tanhf)
    return __builtin_amdgcn_tanhf(x);
#elif __has_builtin(__builtin_amdgcn_tanh_f32)
    return __builtin_amdgcn_tanh_f32(x);
#else
    // Branch-free: tanh(x) = 1 - 2/(1 + 2^(2*log2e*x)).
    // Limits: x->+inf => exp2->inf => 1; x->-inf => exp2->0 => -1.  Two TRANS ops.
    float e = __builtin_amdgcn_exp2f(x * 2.8853900817779268f); // 2*log2(e)
    return 1.0f - 2.0f * __builtin_amdgcn_rcpf(1.0f + e);
#endif
}

// ---------------------------------------------------------------------------
// Projection GEMM via V_WMMA_F32_16X16X4_F32 (full f32 precision).
// out[l,o] = sum_h X[l,h] * W[o*ldw + h]  (+ bias[o] if hasBias)
// One wave per 16x16 output tile; K=1024 swept 4 at a time.
// A-frag (16x4 f32, ISA layout): lane L holds {A[L&15,k+koff], A[L&15,k+koff+1]},
//   koff = (L>=16)?2:0.  B-frag (4x16) mirrors with W rows o0+(L&15).
// C/D (16x16 f32): VGPR r -> row r (lanes 0-15) / row r+8 (lanes 16-31), col=lane&15.
// ---------------------------------------------------------------------------
__global__ __launch_bounds__(256)
void proj_wmma_f32(const float* __restrict__ X, const float* __restrict__ W,
                   const float* __restrict__ bias, float* __restrict__ out,
                   int ldw, int hasBias) {
    const int K = HDIM, N = HDIM;
    int wave  = (blockIdx.x * blockDim.x + threadIdx.x) >> 5; // global tile id
    int lane  = threadIdx.x & 31;
    int tileM = wave >> 6;   // N/16 = 64 tiles per output row-band
    int tileN = wave & 63;
    int l0 = tileM << 4;
    int o0 = tileN << 4;

    int row  = lane & 15;
    int koff = (lane >> 4) << 1;
    const float* aPtr = X + (size_t)(l0 + row) * K   + koff;
    const float* bPtr = W + (size_t)(o0 + row) * ldw + koff;

    v8f acc = {};
#pragma unroll 8
    for (int k0 = 0; k0 < K; k0 += 4) {
        v2f a = *(const v2f*)(aPtr + k0);
        v2f b = *(const v2f*)(bPtr + k0);
        acc = __builtin_amdgcn_wmma_f32_16x16x4_f32(
            /*neg_a=*/false, a, /*neg_b=*/false, b,
            /*c_mod=*/(short)0, acc, /*reuse_a=*/false, /*reuse_b=*/false);
    }

    int col   = lane & 15;
    int rbase = (lane >> 4) << 3;
    float bv  = hasBias ? bias[o0 + col] : 0.0f;
#pragma unroll
    for (int r = 0; r < 8; ++r) {
        out[(size_t)(l0 + rbase + r) * N + o0 + col] = acc[r] + bv;
    }
}

// ---------------------------------------------------------------------------
// Pairwise stage: raw[i,j,c] = sum_h W2[c,h] * tanh(A[j,h] + R[i,h])
// where A = a_proj, R = b_proj + b1 (bias folded in by proj kernel).
// Block = 256 threads handles a 16(i) x 16(j) pair tile; h chunked by HC.
// LDS rows padded to LDSP floats so the 16 distinct tj lanes hit 16 banks.
// ---------------------------------------------------------------------------
__global__ __launch_bounds__(256)
void pairwise_tanh(const float* __restrict__ A, const float* __restrict__ R,
                   const float* __restrict__ W2, float* __restrict__ raw) {
    __shared__ float aT[16][LDSP];
    __shared__ float rT[16][LDSP];
    __shared__ float w2s[2][HC];

    const int i0 = blockIdx.y << 4;
    const int j0 = blockIdx.x << 4;
    const int tid = threadIdx.x;
    const int tj = tid & 15;
    const int ti = tid >> 4;

    float acc0 = 0.0f, acc1 = 0.0f;

    for (int h0 = 0; h0 < HDIM; h0 += HC) {
        __syncthreads();
        // Stage 16 x HC floats of A and R (float4 per thread, 4 iterations)
#pragma unroll
        for (int p = tid; p < 16 * (HC / 4); p += 256) {
            int rrow = p / (HC / 4);
            int c4   = (p % (HC / 4)) << 2;
            float4 va = *(const float4*)(A + (size_t)(j0 + rrow) * HDIM + h0 + c4);
            *(float4*)&aT[rrow][c4] = va;
            float4 vr = *(const float4*)(R + (size_t)(i0 + rrow) * HDIM + h0 + c4);
            *(float4*)&rT[rrow][c4] = vr;
        }
        // Stage W2 chunk (C=2); HC==256 => exactly one element per thread
        {
            int p = tid;
            w2s[0][p] = W2[h0 + p];
            w2s[1][p] = W2[HDIM + h0 + p];
        }
        __syncthreads();

#pragma unroll 4
        for (int hc = 0; hc < HC; ++hc) {
            float x = aT[tj][hc] + rT[ti][hc];
            float t = fast_tanh(x);
            acc0 = fmaf(w2s[0][hc], t, acc0);
            acc1 = fmaf(w2s[1][hc], t, acc1);
        }
    }

    int i = i0 + ti, j = j0 + tj;
    float2 o; o.x = acc0; o.y = acc1;
    *(float2*)(raw + ((size_t)(i * LSEQ + j) << 1)) = o;
}

// ---------------------------------------------------------------------------
// out[i,j,c] = 0.5*(raw[i,j,c] + raw[j,i,c]) + b2[c]
// ---------------------------------------------------------------------------
__global__ __launch_bounds__(256)
void symmetrize(const float* __restrict__ raw, const float* __restrict__ b2,
                float* __restrict__ out) {
    int idx = blockIdx.x * 256 + threadIdx.x;      // over 512*512 pairs
    if (idx >= LSEQ * LSEQ) return;
    int i = idx >> 9, j = idx & (LSEQ - 1);
    float2 rij = *(const float2*)(raw + ((size_t)(i * LSEQ + j) << 1));
    float2 rji = *(const float2*)(raw + ((size_t)(j * LSEQ + i) << 1));
    float2 o;
    o.x = 0.5f * (rij.x + rji.x) + b2[0];
    o.y = 0.5f * (rij.y + rji.y) + b2[1];
    *(float2*)(out + ((size_t)idx << 1)) = o;
}

extern "C" void kernel_launch(void* const* d_in, const int* in_sizes, int n_in,
                              void* d_out, int out_size, void* d_ws, size_t ws_size,
                              hipStream_t stream) {
    const float* X  = (const float*)d_in[0]; // lm_output [512,1024]
    const float* W1 = (const float*)d_in[1]; // [1024, 2048], row stride 2048
    const float* b1 = (const float*)d_in[2]; // [1024]
    const float* W2 = (const float*)d_in[3]; // [2, 1024]
    const float* b2 = (const float*)d_in[4]; // [2]
    float* out = (float*)d_out;              // [512,512,2]

    float* a_ws = (float*)d_ws;                     // 512*1024 f32 = 2MB
    float* r_ws = a_ws + (size_t)LSEQ * HDIM;       // 512*1024 f32 = 2MB (b_proj + b1)
    float* raw  = r_ws + (size_t)LSEQ * HDIM;       // 512*512*2 f32 = 2MB

    // 2048 tiles of 16x16, 8 waves per 256-thread block -> 256 blocks
    proj_wmma_f32<<<256, 256, 0, stream>>>(X, W1,          nullptr, a_ws, 2 * HDIM, 0);
    proj_wmma_f32<<<256, 256, 0, stream>>>(X, W1 + HDIM,   b1,      r_ws, 2 * HDIM, 1);

    dim3 grid(LSEQ / 16, LSEQ / 16); // 32 x 32 pair tiles
    pairwise_tanh<<<grid, 256, 0, stream>>>(a_ws, r_ws, W2, raw);

    symmetrize<<<(LSEQ * LSEQ) / 256, 256, 0, stream>>>(raw, b2, out);
}